// EquivariantSelfAttention_28321014350231
// MI455X (gfx1250) — compile-verified
//
#include <hip/hip_runtime.h>
#include <hip/hip_bf16.h>
#include <math.h>

#define BB 2
#define NN 2048
#define HH 256
#define NHD 8
#define DD 32
#define WIN 32
#define WLEN 65
#define BN (BB*NN)   // 4096
#define BHD (BB*NHD) // 16

typedef __attribute__((ext_vector_type(2))) float v2f;
typedef __attribute__((ext_vector_type(8))) float v8f;
typedef __attribute__((ext_vector_type(4))) unsigned int u32x4;
typedef __attribute__((ext_vector_type(8))) int i32x8;
typedef __attribute__((ext_vector_type(4))) int i32x4;

// ---------------------------------------------------------------------------
// Tensor Data Mover: DMA a (up to 3D) fp32 tile from global to LDS.
// D# packing per CDNA5 ISA §8.3-8.6: group0 = {count, lds_addr, global_addr,
// type=2}; group1 = {data_size=4B, tensor dims, tile dims, 48-bit strides};
// group2 word0 = tensor_dim2 (Z extent). Tracked on TENSORcnt.
// ---------------------------------------------------------------------------
__device__ __forceinline__ void tdm_load_to_lds(
    unsigned lds_off, const void* gptr,
    unsigned tensor_d0, unsigned tensor_d1, unsigned tensor_d2,
    unsigned tile_d0, unsigned tile_d1, unsigned tile_d2,
    unsigned long long stride0, unsigned long long stride1) {
  unsigned long long ga = (unsigned long long)(uintptr_t)gptr;
  u32x4 g0;
  g0[0] = 1u;                                        // count=1, user mode
  g0[1] = lds_off;                                   // LDS byte address
  g0[2] = (unsigned)(ga & 0xFFFFFFFFu);              // global_addr[31:0]
  g0[3] = (unsigned)((ga >> 32) & 0x1FFFFFFu) | (2u << 30);  // [56:32] | type=2
  i32x8 g1;
  g1[0] = 0x00020000;                                // data_size=2 (4 bytes)
  g1[1] = (int)((tensor_d0 & 0xFFFFu) << 16);        // tensor_dim0[15:0] @ b48
  g1[2] = (int)(((tensor_d0 >> 16) & 0xFFFFu) | ((tensor_d1 & 0xFFFFu) << 16));
  g1[3] = (int)(((tensor_d1 >> 16) & 0xFFFFu) | ((tile_d0 & 0xFFFFu) << 16));
  g1[4] = (int)((tile_d1 & 0xFFFFu) | ((tile_d2 & 0xFFFFu) << 16));
  g1[5] = (int)(stride0 & 0xFFFFFFFFull);            // tensor_dim0_stride[31:0]
  g1[6] = (int)(((stride0 >> 32) & 0xFFFFull) | ((stride1 & 0xFFFFull) << 16));
  g1[7] = (int)((stride1 >> 16) & 0xFFFFFFFFull);    // tensor_dim1_stride[47:16]
  i32x4 g2;
  g2[0] = (int)tensor_d2;                            // tensor_dim2
  g2[1] = 0; g2[2] = 0; g2[3] = 0;
  i32x4 g3 = {0, 0, 0, 0};
#if __clang_major__ >= 23
  i32x8 g4 = {0, 0, 0, 0, 0, 0, 0, 0};
  __builtin_amdgcn_tensor_load_to_lds(g0, g1, g2, g3, g4, 0);
#else
  __builtin_amdgcn_tensor_load_to_lds(g0, g1, g2, g3, 0);
#endif
}

// 16x16 fp32 output tile: C = A(16xK, LDS) * B(KxN slice, global), via
// V_WMMA_F32_16X16X4_F32. Lanes 0-15 carry K-offsets {k,k+1}, lanes 16-31
// carry {k+2,k+3} (ISA §7.12.2 32-bit A 16x4 layout; B mirrored).
__device__ __forceinline__ v8f wmma_gemm16(const float* __restrict__ As, int lda,
                                           const float* __restrict__ Bw, int ldb,
                                           int K, int lane) {
  const int half = lane >> 4;
  const int idx  = lane & 15;
  v8f acc = {};
  for (int k = 0; k < K; k += 4) {
    v2f a, b;
    a[0] = As[idx * lda + k + 2*half + 0];
    a[1] = As[idx * lda + k + 2*half + 1];
    b[0] = Bw[(size_t)(k + 2*half + 0) * ldb + idx];
    b[1] = Bw[(size_t)(k + 2*half + 1) * ldb + idx];
    acc = __builtin_amdgcn_wmma_f32_16x16x4_f32(false, a, false, b, (short)0, acc,
                                                false, false);
  }
  return acc;
}

// ---------- Kernel 1: q/k/v projections, output in head-major (BH,N,d) ----------
__global__ void qkv_proj_kernel(const float* __restrict__ x,
                                const float* __restrict__ Wq, const float* __restrict__ bq,
                                const float* __restrict__ Wk, const float* __restrict__ bk,
                                const float* __restrict__ Wv, const float* __restrict__ bv,
                                float* __restrict__ qo, float* __restrict__ ko,
                                float* __restrict__ vo) {
  __shared__ float As[16 * HH];
  const int m0 = blockIdx.x * 16;
  const int tid = threadIdx.x;
  if (tid < 32) {                        // wave 0 drives the TDM
    // 2D tile: 16 rows (stride 1024 floats = x row pitch) x 256 contiguous
    tdm_load_to_lds((unsigned)(uintptr_t)As, x + (size_t)m0 * (4 * HH),
                    /*t_d0=*/HH, /*t_d1=*/16, /*t_d2=*/0,
                    /*tile*/ HH, 16, 0,
                    /*stride0=*/4 * HH, /*stride1=*/0);
    __builtin_amdgcn_s_wait_tensorcnt(0);
  }
  __syncthreads();
  const int wave = tid >> 5, lane = tid & 31;
  const int tileIdx = blockIdx.y * 8 + wave;        // 0..47
  const int mat = tileIdx >> 4;
  const int n0c = (tileIdx & 15) * 16;
  const float* W    = (mat == 0) ? Wq : (mat == 1) ? Wk : Wv;
  const float* bias = (mat == 0) ? bq : (mat == 1) ? bk : bv;
  float* outp       = (mat == 0) ? qo : (mat == 1) ? ko : vo;
  __builtin_prefetch(W + n0c, 0, 1);
  v8f acc = wmma_gemm16(As, HH, W + n0c, HH, HH, lane);
  const int half = lane >> 4, idx = lane & 15;
  const int colg = n0c + idx;
  const int head = colg >> 5, dim = colg & 31;
  const float bb = bias[colg];
  #pragma unroll
  for (int r = 0; r < 8; ++r) {
    int rr = m0 + half * 8 + r;
    int b = rr >> 11, n = rr & (NN - 1);
    outp[(((size_t)(b * NHD + head)) * NN + n) * DD + dim] = acc[r] + bb;
  }
}

// ---------- Kernel 2: vec_dot, fusing (B,N,3,2H) vec_proj entirely in regs ----------
// One 3D TDM: X=256 cols, Y=3 channels (stride 256), Z=16 rows (stride 1024).
// LDS linear order -> [row][a][c], so slice a is As+a*256 with lda=768.
__global__ void vecdot_kernel(const float* __restrict__ x,
                              const float* __restrict__ Wvec,
                              float* __restrict__ vec_dot) {
  __shared__ float As[16 * 3 * HH];
  const int m0 = blockIdx.x * 16;
  const int tid = threadIdx.x;
  if (tid < 32) {
    tdm_load_to_lds((unsigned)(uintptr_t)As, x + (size_t)m0 * (4 * HH) + HH,
                    /*t_d0=*/HH, /*t_d1=*/3, /*t_d2=*/16,
                    /*tile*/ HH, 3, 16,
                    /*stride0=*/HH, /*stride1=*/4 * HH);
    __builtin_amdgcn_s_wait_tensorcnt(0);
  }
  __syncthreads();
  const int wave = tid >> 5, lane = tid & 31;
  const int j0 = (blockIdx.y * 8 + wave) * 16;      // 0..240
  __builtin_prefetch(Wvec + j0, 0, 1);
  v8f vd = {};
  #pragma unroll
  for (int a = 0; a < 3; ++a) {
    v8f c1 = wmma_gemm16(As + a * HH, 3 * HH, Wvec + j0, 2 * HH, HH, lane);
    v8f c2 = wmma_gemm16(As + a * HH, 3 * HH, Wvec + HH + j0, 2 * HH, HH, lane);
    vd += c1 * c2;
  }
  const int half = lane >> 4, idx = lane & 15;
  #pragma unroll
  for (int r = 0; r < 8; ++r) {
    int rr = m0 + half * 8 + r;
    vec_dot[(size_t)rr * HH + j0 + idx] = vd[r];
  }
}

// ---------- Kernel 3: vec_norm over the 3 vector channels ----------
__global__ void vecnorm_kernel(const float* __restrict__ x, float* __restrict__ vec_norm) {
  int i = blockIdx.x * blockDim.x + threadIdx.x;
  if (i >= BN * HH) return;
  int r = i >> 8, c = i & 255;
  float v1 = x[((size_t)r * 4 + 1) * HH + c];
  float v2 = x[((size_t)r * 4 + 2) * HH + c];
  float v3 = x[((size_t)r * 4 + 3) * HH + c];
  vec_norm[i] = sqrtf(v1 * v1 + v2 * v2 + v3 * v3);
}

// ---------- Kernel 4: gate = sigmoid([a_d*dot, a_n*norm] @ Wg + bg) ----------
__global__ void gate_kernel(const float* __restrict__ vec_dot, const float* __restrict__ vec_norm,
                            const float* __restrict__ alpha_dot, const float* __restrict__ alpha_norm,
                            const float* __restrict__ Wg, const float* __restrict__ bg,
                            float* __restrict__ gate) {
  __shared__ float As[16 * 2 * HH];
  const int m0 = blockIdx.x * 16;
  const int tid = threadIdx.x;
  const float ad = alpha_dot[0], an = alpha_norm[0];
  for (int e = tid; e < 16 * 2 * HH; e += 256) {   // computed tile: stays manual
    int row = e >> 9, c = e & 511;
    As[e] = (c < HH) ? ad * vec_dot[(size_t)(m0 + row) * HH + c]
                     : an * vec_norm[(size_t)(m0 + row) * HH + (c - HH)];
  }
  __syncthreads();
  const int wave = tid >> 5, lane = tid & 31;
  const int j0 = (blockIdx.y * 8 + wave) * 16;
  __builtin_prefetch(Wg + j0, 0, 1);
  v8f acc = wmma_gemm16(As, 2 * HH, Wg + j0, HH, 2 * HH, lane);
  const int half = lane >> 4, idx = lane & 15;
  const float bgv = bg[j0 + idx];
  #pragma unroll
  for (int r = 0; r < 8; ++r) {
    int rr = m0 + half * 8 + r;
    float z = acc[r] + bgv;
    gate[(size_t)rr * HH + j0 + idx] = 1.0f / (1.0f + __expf(-z));
  }
}

// ---------- Kernel 5: sliding-window attention; lane == head-dim (d=32 = wave32) ----------
#define TAT 16
#define ROWS (TAT + 2 * WIN)   // 80
__global__ void attn_kernel(const float* __restrict__ q, const float* __restrict__ k,
                            const float* __restrict__ v, const float* __restrict__ x,
                            float* __restrict__ x_agg, float* __restrict__ vec_aggr,
                            float* __restrict__ attn_w) {
  __shared__ float lk[ROWS * 33];        // padded stride: kill bank conflicts
  __shared__ float lv[ROWS * DD];
  __shared__ float lvec[ROWS * 96];
  __shared__ float lq[TAT * DD];
  const int bh = blockIdx.y;
  const int b = bh >> 3, head = bh & 7;
  const int n0 = blockIdx.x * TAT;
  const int tid = threadIdx.x;
  for (int e = tid; e < ROWS * 160; e += 256) {    // halo zero-fill: stays manual
    int row = e / 160, c = e % 160;
    int j = n0 - WIN + row;
    bool ok = (j >= 0) && (j < NN);
    if (c < 32) {
      lk[row * 33 + c] = ok ? k[((size_t)bh * NN + j) * DD + c] : 0.0f;
    } else if (c < 64) {
      lv[row * DD + (c - 32)] = ok ? v[((size_t)bh * NN + j) * DD + (c - 32)] : 0.0f;
    } else {
      int vc = c - 64;
      int a = vc >> 5, dim = vc & 31;
      lvec[row * 96 + vc] =
          ok ? x[(((size_t)(b * NN + j)) * 4 + 1 + a) * HH + head * DD + dim] : 0.0f;
    }
  }
  for (int e = tid; e < TAT * DD; e += 256) {
    lq[e] = q[((size_t)bh * NN + n0 + (e >> 5)) * DD + (e & 31)];
  }
  __syncthreads();
  const int wave = tid >> 5, lane = tid & 31;
  const float scale = 0.17677669529663687f;   // 1/sqrt(32)
  for (int p = 0; p < 2; ++p) {
    const int pr = wave * 2 + p;
    const int n = n0 + pr;
    float s[3], e_[3];
    float mloc = -1e30f;
    #pragma unroll
    for (int t = 0; t < 3; ++t) {            // lane owns window offsets lane, lane+32, lane+64
      int wj = lane + t * 32;
      float sc = -1e30f;
      if (wj < WLEN) {
        int row = pr + wj;
        float acc = 0.0f;
        #pragma unroll
        for (int dd2 = 0; dd2 < DD; ++dd2)
          acc += lq[pr * DD + dd2] * lk[row * 33 + dd2];
        sc = acc * scale;
      }
      s[t] = sc;
      mloc = fmaxf(mloc, sc);
    }
    for (int off = 16; off > 0; off >>= 1)
      mloc = fmaxf(mloc, __shfl_xor(mloc, off, 32));
    float ssum = 0.0f;
    #pragma unroll
    for (int t = 0; t < 3; ++t) {
      int wj = lane + t * 32;
      float ev = (wj < WLEN) ? __expf(s[t] - mloc) : 0.0f;
      e_[t] = ev;
      ssum += ev;
    }
    for (int off = 16; off > 0; off >>= 1)
      ssum += __shfl_xor(ssum, off, 32);
    const float inv = 1.0f / ssum;
    float wreg[3] = {e_[0] * inv, e_[1] * inv, e_[2] * inv};
    size_t abase = ((size_t)bh * NN + n) * WLEN;
    attn_w[abase + lane]      = wreg[0];
    attn_w[abase + lane + 32] = wreg[1];
    if (lane == 0) attn_w[abase + 64] = wreg[2];
    float ax = 0.f, a0 = 0.f, a1 = 0.f, a2 = 0.f;
    #pragma unroll
    for (int t = 0; t < 3; ++t) {
      const int lim = (t < 2) ? 32 : 1;
      for (int l = 0; l < lim; ++l) {
        float wt = __shfl(wreg[t], l, 32);   // broadcast weight to all lanes
        int row = pr + t * 32 + l;
        ax += wt * lv[row * DD + lane];
        a0 += wt * lvec[row * 96 + lane];
        a1 += wt * lvec[row * 96 + 32 + lane];
        a2 += wt * lvec[row * 96 + 64 + lane];
      }
    }
    size_t rb = (size_t)(b * NN + n);
    x_agg[rb * HH + head * DD + lane] = ax;                  // (B,N,H) fused transpose
    vec_aggr[(rb * 3 + 0) * HH + head * DD + lane] = a0;     // (B,N,3,H)
    vec_aggr[(rb * 3 + 1) * HH + head * DD + lane] = a1;
    vec_aggr[(rb * 3 + 2) * HH + head * DD + lane] = a2;
  }
}

// ---------- Kernel 6: o = x_out @ Wo + bo ----------
__global__ void o_proj_kernel(const float* __restrict__ xagg, const float* __restrict__ Wo,
                              const float* __restrict__ bo, float* __restrict__ o) {
  __shared__ float As[16 * HH];
  const int m0 = blockIdx.x * 16;
  const int tid = threadIdx.x;
  if (tid < 32) {
    tdm_load_to_lds((unsigned)(uintptr_t)As, xagg + (size_t)m0 * HH,
                    /*t_d0=*/HH, /*t_d1=*/16, /*t_d2=*/0,
                    /*tile*/ HH, 16, 0,
                    /*stride0=*/HH, /*stride1=*/0);
    __builtin_amdgcn_s_wait_tensorcnt(0);
  }
  __syncthreads();
  const int wave = tid >> 5, lane = tid & 31;
  const int j0 = (blockIdx.y * 8 + wave) * 16;   // 0..752
  __builtin_prefetch(Wo + j0, 0, 1);
  v8f acc = wmma_gemm16(As, HH, Wo + j0, 3 * HH, HH, lane);
  const int half = lane >> 4, idx = lane & 15;
  const float bcol = bo[j0 + idx];
  #pragma unroll
  for (int r = 0; r < 8; ++r) {
    int rr = m0 + half * 8 + r;
    o[(size_t)rr * (3 * HH) + j0 + idx] = acc[r] + bcol;
  }
}

// ---------- Kernel 7: finalize x_updated + gated vec combine ----------
__global__ void finalize_kernel(const float* __restrict__ x, const float* __restrict__ o,
                                const float* __restrict__ vec_dot, const float* __restrict__ vec_norm,
                                const float* __restrict__ gate, const float* __restrict__ vec_aggr,
                                float* __restrict__ x_final) {
  int i = blockIdx.x * blockDim.x + threadIdx.x;
  if (i >= BN * HH) return;
  int r = i >> 8, c = i & 255;
  float vd = vec_dot[i], vn = vec_norm[i], g = gate[i];
  float o1 = o[(size_t)r * 768 + c];
  float o2 = o[(size_t)r * 768 + 256 + c];
  float o3 = o[(size_t)r * 768 + 512 + c];
  x_final[(size_t)r * 1024 + c] = vd * o1 + vn * o2 + o3;
  #pragma unroll
  for (int a = 0; a < 3; ++a) {
    float vecv = x[(size_t)r * 1024 + (1 + a) * 256 + c];
    float va = vec_aggr[((size_t)r * 3 + a) * 256 + c];
    x_final[(size_t)r * 1024 + (1 + a) * 256 + c] = g * va + vecv;
  }
}

extern "C" void kernel_launch(void* const* d_in, const int* in_sizes, int n_in,
                              void* d_out, int out_size, void* d_ws, size_t ws_size,
                              hipStream_t stream) {
  (void)in_sizes; (void)n_in; (void)out_size; (void)ws_size;
  const float* x    = (const float*)d_in[0];
  const float* Wq   = (const float*)d_in[1];
  const float* bq   = (const float*)d_in[2];
  const float* Wk   = (const float*)d_in[3];
  const float* bk   = (const float*)d_in[4];
  const float* Wv   = (const float*)d_in[5];
  const float* bv   = (const float*)d_in[6];
  const float* Wo   = (const float*)d_in[7];
  const float* bo   = (const float*)d_in[8];
  const float* Wvec = (const float*)d_in[9];
  const float* ad   = (const float*)d_in[10];
  const float* an   = (const float*)d_in[11];
  const float* Wg   = (const float*)d_in[12];
  const float* bg   = (const float*)d_in[13];

  float* ws = (float*)d_ws;
  float* qb       = ws;                    // 16*2048*32 = 1,048,576
  float* kb       = qb + 1048576;
  float* vb       = kb + 1048576;
  float* vec_dot  = vb + 1048576;          // 4096*256
  float* vec_norm = vec_dot + 1048576;
  float* gate     = vec_norm + 1048576;
  float* x_agg    = gate + 1048576;
  float* vec_aggr = x_agg + 1048576;       // 4096*3*256
  float* ob       = vec_aggr + 3145728;    // 4096*768

  float* x_final = (float*)d_out;
  float* attn_w  = x_final + (size_t)BB * NN * 4 * HH;   // + 4,194,304

  qkv_proj_kernel<<<dim3(BN / 16, 6), 256, 0, stream>>>(x, Wq, bq, Wk, bk, Wv, bv, qb, kb, vb);
  vecdot_kernel  <<<dim3(BN / 16, 2), 256, 0, stream>>>(x, Wvec, vec_dot);
  vecnorm_kernel <<<(BN * HH) / 256, 256, 0, stream>>>(x, vec_norm);
  gate_kernel    <<<dim3(BN / 16, 2), 256, 0, stream>>>(vec_dot, vec_norm, ad, an, Wg, bg, gate);
  attn_kernel    <<<dim3(NN / TAT, BHD), 256, 0, stream>>>(qb, kb, vb, x, x_agg, vec_aggr, attn_w);
  o_proj_kernel  <<<dim3(BN / 16, 6), 256, 0, stream>>>(x_agg, Wo, bo, ob);
  finalize_kernel<<<(BN * HH) / 256, 256, 0, stream>>>(x, ob, vec_dot, vec_norm, gate, vec_aggr, x_final);
}